// ForwardNetwork_49761491091770
// MI455X (gfx1250) — compile-verified
//
#include <hip/hip_runtime.h>
#include <math.h>

typedef __attribute__((ext_vector_type(2))) float v2f;
typedef __attribute__((ext_vector_type(8))) float v8f;
typedef __attribute__((ext_vector_type(4))) int   i4;
typedef __attribute__((ext_vector_type(4))) float f4;

#define THREADS 256

// ---------------------------------------------------------------------------
// K0: deg[i] = 1.0f (self-loop contribution); zero the 4-float mean accumulator
// ---------------------------------------------------------------------------
__global__ void k0_init(float* __restrict__ degbuf, float* __restrict__ gsum, int n) {
  int i = blockIdx.x * blockDim.x + threadIdx.x;
  if (i < n) degbuf[i] = 1.0f;
  if (i < 4) gsum[i] = 0.0f;
}

// ---------------------------------------------------------------------------
// K1: degree accumulation over col (target) indices. Edge list is read-once:
// use non-temporal int4 loads so it does not evict the hot accumulator in L2.
// ---------------------------------------------------------------------------
__global__ void k1_degree(const int* __restrict__ col, float* __restrict__ degbuf, int e) {
  const int stride = gridDim.x * blockDim.x;
  const int tid = blockIdx.x * blockDim.x + threadIdx.x;
  const int e4 = e >> 2;
  const i4* col4 = (const i4*)col;
  for (int i = tid; i < e4; i += stride) {
    i4 c = __builtin_nontemporal_load(&col4[i]);
    atomicAdd(&degbuf[c.x], 1.0f);
    atomicAdd(&degbuf[c.y], 1.0f);
    atomicAdd(&degbuf[c.z], 1.0f);
    atomicAdd(&degbuf[c.w], 1.0f);
  }
  for (int i = (e4 << 2) + tid; i < e; i += stride)
    atomicAdd(&degbuf[col[i]], 1.0f);
}

// ---------------------------------------------------------------------------
// K2a: dinv[i] = deg[i]^-0.5 in place (deg >= 1 always, self-loop).
// ---------------------------------------------------------------------------
__global__ void k2a_rsqrt(float* __restrict__ degbuf, int n) {
  int i = blockIdx.x * blockDim.x + threadIdx.x;
  if (i < n) degbuf[i] = rsqrtf(degbuf[i]);
}

// ---------------------------------------------------------------------------
// K2b: hs[i] = dinv[i] * (x[i] @ conv_w^T), via V_WMMA_F32_16X16X4_F32.
// One wave handles a 16-node tile: D(16x16) = A(16x4) x B(4x16) + C,
// K=6 split into chunks {0..3} and {4,5 + zero pad}.
//   A layout (32-bit 16x4): lanes 0-15: v0=K0,v1=K1 ; lanes 16-31: v0=K2,v1=K3
//   B layout (32-bit 4x16): lane = N (mod 16); lanes<16: v0=K0,v1=K1 ; >=16: K2,K3
//   D layout (32-bit 16x16): VGPR v: lane<16 -> (M=v, N=lane) ; lane>=16 -> (M=v+8)
// Also seeds S with the self-loop term (S[i] = hs[i]).
// ---------------------------------------------------------------------------
__global__ void k2b_conv_wmma(const float* __restrict__ x,
                              const float* __restrict__ conv_w,
                              const float* __restrict__ dinv,
                              float* __restrict__ hs,
                              float* __restrict__ S,
                              int n) {
  const int lane = threadIdx.x & 31;
  const int wave = (blockIdx.x * blockDim.x + threadIdx.x) >> 5;
  const int nwaves = (gridDim.x * blockDim.x) >> 5;
  const int ntiles = (n + 15) >> 4;

  const int  nl = lane & 15;        // node-within-tile (A rows) / column N (B,D)
  const bool lo = lane < 16;
  const int  kbase = lo ? 0 : 2;

  // B = conv_w^T (6x4), padded to two 4x16 chunks. conv_w is [4,6] row-major:
  // W^T[k][nn] = conv_w[nn*6 + k]. Columns nn >= 4 are zero.
  v2f b0 = {0.f, 0.f}, b1 = {0.f, 0.f};
  if (nl < 4) {
    b0.x = conv_w[nl * 6 + kbase];
    b0.y = conv_w[nl * 6 + kbase + 1];
    if (lo) { b1.x = conv_w[nl * 6 + 4]; b1.y = conv_w[nl * 6 + 5]; }
  }

  for (int t = wave; t < ntiles; t += nwaves) {
    const int base = t << 4;
    const int node = base + nl;
    const int ldn  = node < n ? node : n - 1;

    v2f a0, a1;
    a0.x = x[ldn * 6 + kbase];
    a0.y = x[ldn * 6 + kbase + 1];
    a1.x = lo ? x[ldn * 6 + 4] : 0.f;
    a1.y = lo ? x[ldn * 6 + 5] : 0.f;
    if (node >= n) { a0.x = 0.f; a0.y = 0.f; a1.x = 0.f; a1.y = 0.f; }

    v8f c = {0.f, 0.f, 0.f, 0.f, 0.f, 0.f, 0.f, 0.f};
    c = __builtin_amdgcn_wmma_f32_16x16x4_f32(false, a0, false, b0,
                                              (short)0, c, false, false);
    c = __builtin_amdgcn_wmma_f32_16x16x4_f32(false, a1, false, b1,
                                              (short)0, c, false, false);

    if (nl < 4) {                        // only N=0..3 columns carry outputs
      const int mofs = lo ? 0 : 8;
#pragma unroll
      for (int v = 0; v < 8; ++v) {
        const int m = base + mofs + v;
        if (m < n) {
          const float val = c[v] * dinv[m];
          hs[(size_t)m * 4 + nl] = val;   // immutable gather source
          S [(size_t)m * 4 + nl] = val;   // accumulator seeded with self-loop
        }
      }
    }
  }
}

// ---------------------------------------------------------------------------
// K3: edge scatter-add  S[col] += hs[row]  (float4 per edge, 4 atomics).
// Edge lists streamed non-temporally; hs/S stay resident in L2.
// ---------------------------------------------------------------------------
__global__ void k3_scatter(const int* __restrict__ row, const int* __restrict__ col,
                           const float* __restrict__ hs, float* __restrict__ S, int e) {
  const int stride = gridDim.x * blockDim.x;
  const int tid = blockIdx.x * blockDim.x + threadIdx.x;
  const int e4 = e >> 2;
  const i4* row4 = (const i4*)row;
  const i4* col4 = (const i4*)col;
  const f4* hs4  = (const f4*)hs;
  for (int i = tid; i < e4; i += stride) {
    i4 r = __builtin_nontemporal_load(&row4[i]);
    i4 c = __builtin_nontemporal_load(&col4[i]);
    f4 m0 = hs4[r.x];
    atomicAdd(&S[(size_t)c.x * 4 + 0], m0.x); atomicAdd(&S[(size_t)c.x * 4 + 1], m0.y);
    atomicAdd(&S[(size_t)c.x * 4 + 2], m0.z); atomicAdd(&S[(size_t)c.x * 4 + 3], m0.w);
    f4 m1 = hs4[r.y];
    atomicAdd(&S[(size_t)c.y * 4 + 0], m1.x); atomicAdd(&S[(size_t)c.y * 4 + 1], m1.y);
    atomicAdd(&S[(size_t)c.y * 4 + 2], m1.z); atomicAdd(&S[(size_t)c.y * 4 + 3], m1.w);
    f4 m2 = hs4[r.z];
    atomicAdd(&S[(size_t)c.z * 4 + 0], m2.x); atomicAdd(&S[(size_t)c.z * 4 + 1], m2.y);
    atomicAdd(&S[(size_t)c.z * 4 + 2], m2.z); atomicAdd(&S[(size_t)c.z * 4 + 3], m2.w);
    f4 m3 = hs4[r.w];
    atomicAdd(&S[(size_t)c.w * 4 + 0], m3.x); atomicAdd(&S[(size_t)c.w * 4 + 1], m3.y);
    atomicAdd(&S[(size_t)c.w * 4 + 2], m3.z); atomicAdd(&S[(size_t)c.w * 4 + 3], m3.w);
  }
  for (int i = (e4 << 2) + tid; i < e; i += stride) {
    int r = row[i], c = col[i];
    f4 m = hs4[r];
    atomicAdd(&S[(size_t)c * 4 + 0], m.x); atomicAdd(&S[(size_t)c * 4 + 1], m.y);
    atomicAdd(&S[(size_t)c * 4 + 2], m.z); atomicAdd(&S[(size_t)c * 4 + 3], m.w);
  }
}

// ---------------------------------------------------------------------------
// K4: t = tanh(dinv[i]*S[i] + conv_b); block-reduce Σt and atomically fold
// into the 4-float global accumulator (mean finished in K5).
// ---------------------------------------------------------------------------
__global__ void k4_tanh_reduce(const float* __restrict__ S, const float* __restrict__ dinv,
                               const float* __restrict__ conv_b, float* __restrict__ gsum,
                               int n) {
  __shared__ float red[THREADS];
  const int stride = gridDim.x * blockDim.x;
  const int tid = blockIdx.x * blockDim.x + threadIdx.x;
  const float b0 = conv_b[0], b1 = conv_b[1], b2 = conv_b[2], b3 = conv_b[3];
  float t0 = 0.f, t1 = 0.f, t2 = 0.f, t3 = 0.f;
  const f4* S4 = (const f4*)S;
  for (int j = tid; j < n; j += stride) {
    float dv = dinv[j];
    f4 s = S4[j];
    t0 += tanhf(fmaf(dv, s.x, b0));
    t1 += tanhf(fmaf(dv, s.y, b1));
    t2 += tanhf(fmaf(dv, s.z, b2));
    t3 += tanhf(fmaf(dv, s.w, b3));
  }
  float tv[4] = {t0, t1, t2, t3};
#pragma unroll
  for (int k = 0; k < 4; ++k) {
    red[threadIdx.x] = tv[k];
    __syncthreads();
    for (int ofs = blockDim.x >> 1; ofs > 0; ofs >>= 1) {
      if (threadIdx.x < (unsigned)ofs) red[threadIdx.x] += red[threadIdx.x + ofs];
      __syncthreads();
    }
    if (threadIdx.x == 0) atomicAdd(&gsum[k], red[0]);
    __syncthreads();
  }
}

// ---------------------------------------------------------------------------
// K5: tiny FC head on one block: g = gsum/n ; hdn = relu(g@fc1^T+b1) ;
// out = hdn@fc3^T+b3  -> 10 floats.
// ---------------------------------------------------------------------------
__global__ void k5_head(const float* __restrict__ gsum,
                        const float* __restrict__ fc1_w, const float* __restrict__ fc1_b,
                        const float* __restrict__ fc3_w, const float* __restrict__ fc3_b,
                        float* __restrict__ out, float inv_n) {
  __shared__ float g4[4];
  __shared__ float hdn[64];
  const int tid = threadIdx.x;
  if (tid < 4) g4[tid] = gsum[tid] * inv_n;
  __syncthreads();
  if (tid < 64) {
    float acc = fc1_b[tid];
#pragma unroll
    for (int k = 0; k < 4; ++k) acc = fmaf(g4[k], fc1_w[tid * 4 + k], acc);
    hdn[tid] = fmaxf(acc, 0.f);
  }
  __syncthreads();
  if (tid < 10) {
    float acc = fc3_b[tid];
#pragma unroll
    for (int j = 0; j < 64; ++j) acc = fmaf(hdn[j], fc3_w[tid * 64 + j], acc);
    out[tid] = acc;
  }
}

// ---------------------------------------------------------------------------
extern "C" void kernel_launch(void* const* d_in, const int* in_sizes, int n_in,
                              void* d_out, int out_size, void* d_ws, size_t ws_size,
                              hipStream_t stream) {
  const float* x      = (const float*)d_in[0];
  const int*   ei     = (const int*)  d_in[1];   // [2, E] int32, flat
  const float* conv_w = (const float*)d_in[2];
  const float* conv_b = (const float*)d_in[3];
  const float* fc1_w  = (const float*)d_in[4];
  const float* fc1_b  = (const float*)d_in[5];
  const float* fc3_w  = (const float*)d_in[6];
  const float* fc3_b  = (const float*)d_in[7];
  float* out = (float*)d_out;

  const int n = in_sizes[0] / 6;
  const int e = in_sizes[1] / 2;
  const int* row = ei;          // edge_index[0] : source
  const int* col = ei + e;      // edge_index[1] : aggregation target

  // Workspace carve-out (256B-aligned regions): dinv[n], hs[4n], S[4n], gsum[4]
  char* ws = (char*)d_ws;
  size_t ofs = 0;
  auto carve = [&](size_t bytes) -> char* {
    char* p = ws + ofs;
    ofs = (ofs + bytes + 255) & ~(size_t)255;
    return p;
  };
  float* degbuf = (float*)carve((size_t)n * sizeof(float));       // deg -> dinv
  float* hs     = (float*)carve((size_t)n * 4 * sizeof(float));
  float* S      = (float*)carve((size_t)n * 4 * sizeof(float));
  float* gsum   = (float*)carve(4 * sizeof(float));
  (void)ws_size; (void)n_in; (void)out_size;

  const int nBlocks  = (n + THREADS - 1) / THREADS;
  const int ntiles   = (n + 15) >> 4;
  const int wmmaBlks = (ntiles + 7) / 8;        // 8 waves / 256-thread block
  const int edgeBlks = 4096;

  k0_init<<<nBlocks, THREADS, 0, stream>>>(degbuf, gsum, n);
  k1_degree<<<edgeBlks, THREADS, 0, stream>>>(col, degbuf, e);
  k2a_rsqrt<<<nBlocks, THREADS, 0, stream>>>(degbuf, n);
  k2b_conv_wmma<<<wmmaBlks, THREADS, 0, stream>>>(x, conv_w, degbuf, hs, S, n);
  k3_scatter<<<edgeBlks * 2, THREADS, 0, stream>>>(row, col, hs, S, e);
  k4_tanh_reduce<<<nBlocks, THREADS, 0, stream>>>(S, degbuf, conv_b, gsum, n);
  k5_head<<<1, 64, 0, stream>>>(gsum, fc1_w, fc1_b, fc3_w, fc3_b, out,
                                1.0f / (float)n);
}